// Rotation_601295421923
// MI455X (gfx1250) — compile-verified
//
#include <hip/hip_runtime.h>

#define DIM    4096
#define KROT   8
#define GROUP  128
#define NG     32      // DIM / GROUP
#define NROWS  16384
#define HALF_G 64      // rotation pairs per group

// LDS strides (in elements) chosen for bank-conflict-free access patterns
#define MSTR 129       // compose kernel: matrix row stride (floats)
#define XSTR 132       // apply kernel: x-tile row stride (floats); A b64 reads hit 64 distinct banks
#define BSTR 80        // apply kernel: B-tile row stride (float2); 80*2=160 floats == 32 mod 64 banks

typedef __attribute__((ext_vector_type(2))) float v2f;
typedef __attribute__((ext_vector_type(8))) float v8f;

// ---------------------------------------------------------------------------
// Kernel A: per group g, compose the 8 Givens layers + channel scaling into a
// single 128x128 matrix M_g with  out_row = in_row * M_g.
// M starts as identity; for each rotation layer k (applied on the right):
//   col_e' = c*col_e - s*col_o ;  col_o' = s*col_e + c*col_o
// Finally each column j is scaled by channel_scales[g*128+j].
// Exported to workspace as float2 "row pairs" so the GEMM kernel's WMMA
// B-fragments are single ds_read_b64:
//   wsp[(g*64 + rp)*128 + j] = ( M[2rp][j], M[2rp+1][j] )
// ---------------------------------------------------------------------------
__global__ __launch_bounds__(256)
void rot_compose(const int* __restrict__ pairs, const float* __restrict__ theta,
                 const float* __restrict__ scales, float2* __restrict__ wsp)
{
    extern __shared__ float M[];             // GROUP * MSTR floats (~66 KB)
    const int g = blockIdx.x;
    const int t = threadIdx.x;

    // identity init
    for (int i = t; i < GROUP * GROUP; i += 256) {
        int r = i >> 7, c = i & 127;
        M[r * MSTR + c] = (r == c) ? 1.0f : 0.0f;
    }
    __syncthreads();

    for (int k = 0; k < KROT; ++k) {
        const int*   pk = pairs + k * DIM + g * GROUP;        // local channel ids 0..127
        const float* tk = theta + k * (DIM / 2) + g * HALF_G; // one angle per pair
        // 64 pairs x 128 rows; pairs are a permutation -> no write conflicts
        for (int i = t; i < HALF_G * GROUP; i += 256) {
            int q = i >> 7;           // pair index in group
            int r = i & 127;          // matrix row
            int e = pk[2 * q];
            int o = pk[2 * q + 1];
            float th = tk[q];
            float cth = cosf(th);
            float sth = sinf(th);
            float a = M[r * MSTR + e];
            float b = M[r * MSTR + o];
            M[r * MSTR + e] = a * cth - b * sth;
            M[r * MSTR + o] = a * sth + b * cth;
        }
        __syncthreads();
    }

    // scale columns, export as (even,odd) row pairs
    const float* sc = scales + g * GROUP;
    for (int i = t; i < HALF_G * GROUP; i += 256) {
        int rp = i >> 7, j = i & 127;
        float s = sc[j];
        float2 v;
        v.x = M[(2 * rp)     * MSTR + j] * s;
        v.y = M[(2 * rp + 1) * MSTR + j] * s;
        wsp[(g * HALF_G + rp) * GROUP + j] = v;
    }
}

// ---------------------------------------------------------------------------
// Kernel B: out[n, g*128+j] = sum_c x[n, g*128+c] * M_g[c][j]
// Block = 256 threads (8 waves), tile = 128 rows x 128 cols (one group).
// Wave w computes rows w*16..w*16+15 for all 128 output cols (8 WMMA col-tiles)
// with V_WMMA_F32_16X16X4_F32, K-loop over 128 in steps of 4.
// Ping-pong double buffering (manual 2x unroll, no rotation copies): each
// WMMA batch runs with exactly one 9-load fragment batch in flight, so the
// scheduler emits partial s_wait_dscnt and DS latency hides under WMMA issue.
//
// Fragment layouts per CDNA5 ISA 7.12.2:
//  A (16x4 f32):  lane L(0-15): M=L, v0=K0, v1=K1 ; lanes 16-31: v0=K2, v1=K3
//  B (4x16 f32):  lane n(0-15): v0=B[0][n], v1=B[1][n]; lanes 16-31: v0=B[2][n], v1=B[3][n]
//  C/D (16x16):   lane n(0-15), VGPR r: D[r][n]; lanes 16-31: D[r+8][n]
// ---------------------------------------------------------------------------
__global__ __launch_bounds__(256)
void rot_apply(const float* __restrict__ x, const float2* __restrict__ wsp,
               float* __restrict__ out)
{
    extern __shared__ unsigned char smem[];
    float*  xs = (float*)smem;                                 // 128 * XSTR floats
    float2* bs = (float2*)(smem + 128 * XSTR * sizeof(float)); // HALF_G * BSTR float2

    const int g    = blockIdx.y;
    const int row0 = blockIdx.x * 128;
    const int t    = threadIdx.x;
    const int lane = t & 31;
    const int w    = t >> 5;

    // --- stage x tile: rows row0..row0+127, cols g*128..g*128+127 (float4 coalesced)
    {
        const float4* xg = (const float4*)(x + (size_t)row0 * DIM + g * GROUP);
        for (int i = t; i < 128 * 32; i += 256) {
            int r  = i >> 5;
            int c4 = i & 31;
            float4 v = xg[(size_t)r * (DIM / 4) + c4];
            float* dst = &xs[r * XSTR + c4 * 4];
            dst[0] = v.x; dst[1] = v.y; dst[2] = v.z; dst[3] = v.w;
        }
    }
    // --- stage M_g row-pairs
    {
        const float2* src = wsp + (size_t)g * HALF_G * GROUP;
        for (int i = t; i < HALF_G * GROUP; i += 256) {
            int p = i >> 7, j = i & 127;
            bs[p * BSTR + j] = src[i];
        }
    }
    __syncthreads();

    v8f acc[8];
#pragma unroll
    for (int jt = 0; jt < 8; ++jt) acc[jt] = v8f{};

    const int arow   = w * 16 + (lane & 15);   // A: M = lane&15
    const int koff   = (lane >> 4) * 2;        // A: upper half-wave holds K2,K3
    const int bp_off = (lane >> 4);            // B: upper half-wave holds K2,K3 pair-row
    const int jn     = lane & 15;              // B/D: N = lane&15

    const float*  aptr = &xs[arow * XSTR + koff];
    const float2* bptr = &bs[bp_off * BSTR + jn];

    // fragment loaders for K-step s (K = 4*s)
#define LOAD_A(s)      (*(const v2f*)(aptr + 4 * (s)))
#define LOAD_B(s, jt)  (*(const v2f*)(bptr + 2 * (s) * BSTR + (jt) * 16))

    // ---- ping-pong pipelined K loop: 32 K-steps of K=4 ----
    v2f aA, aB;
    v2f bA[8], bB[8];

    aA = LOAD_A(0);
#pragma unroll
    for (int jt = 0; jt < 8; ++jt) bA[jt] = LOAD_B(0, jt);

#pragma unroll 1
    for (int i = 0; i < 15; ++i) {
        const int s0 = 2 * i + 1;
        const int s1 = 2 * i + 2;
        // buffer B <- step s0, then consume buffer A (step 2i)
        aB = LOAD_A(s0);
#pragma unroll
        for (int jt = 0; jt < 8; ++jt) bB[jt] = LOAD_B(s0, jt);
#pragma unroll
        for (int jt = 0; jt < 8; ++jt)
            acc[jt] = __builtin_amdgcn_wmma_f32_16x16x4_f32(
                false, aA, false, bA[jt], (short)0, acc[jt], false, false);
        // buffer A <- step s1, then consume buffer B (step s0)
        aA = LOAD_A(s1);
#pragma unroll
        for (int jt = 0; jt < 8; ++jt) bA[jt] = LOAD_B(s1, jt);
#pragma unroll
        for (int jt = 0; jt < 8; ++jt)
            acc[jt] = __builtin_amdgcn_wmma_f32_16x16x4_f32(
                false, aB, false, bB[jt], (short)0, acc[jt], false, false);
    }
    // epilogue: steps 30 (in buffer A) and 31
    aB = LOAD_A(31);
#pragma unroll
    for (int jt = 0; jt < 8; ++jt) bB[jt] = LOAD_B(31, jt);
#pragma unroll
    for (int jt = 0; jt < 8; ++jt)
        acc[jt] = __builtin_amdgcn_wmma_f32_16x16x4_f32(
            false, aA, false, bA[jt], (short)0, acc[jt], false, false);
#pragma unroll
    for (int jt = 0; jt < 8; ++jt)
        acc[jt] = __builtin_amdgcn_wmma_f32_16x16x4_f32(
            false, aB, false, bB[jt], (short)0, acc[jt], false, false);

#undef LOAD_A
#undef LOAD_B

    // --- store D tiles
    const int rbase = row0 + w * 16 + (lane >> 4) * 8;
#pragma unroll
    for (int jt = 0; jt < 8; ++jt) {
        int col = g * GROUP + jt * 16 + jn;
#pragma unroll
        for (int r = 0; r < 8; ++r) {
            out[(size_t)(rbase + r) * DIM + col] = acc[jt][r];
        }
    }
}

// ---------------------------------------------------------------------------
extern "C" void kernel_launch(void* const* d_in, const int* in_sizes, int n_in,
                              void* d_out, int out_size, void* d_ws, size_t ws_size,
                              hipStream_t stream)
{
    const float* xin    = (const float*)d_in[0];
    const int*   pairs  = (const int*)d_in[1];
    const float* theta  = (const float*)d_in[2];
    const float* scales = (const float*)d_in[3];
    float*       outp   = (float*)d_out;
    float2*      wsp    = (float2*)d_ws;   // NG*HALF_G*GROUP float2 = 2 MiB

    const size_t ldsA = (size_t)GROUP * MSTR * sizeof(float);
    const size_t ldsB = (size_t)128 * XSTR * sizeof(float)
                      + (size_t)HALF_G * BSTR * sizeof(float2);

    rot_compose<<<dim3(NG), dim3(256), ldsA, stream>>>(pairs, theta, scales, wsp);
    rot_apply<<<dim3(NROWS / 128, NG), dim3(256), ldsB, stream>>>(xin, wsp, outp);
}